// Lowpass_72980084294307
// MI455X (gfx1250) — compile-verified
//
#include <hip/hip_runtime.h>
#include <cfloat>
#include <cstdint>

// Lowpass IIR scan: y_t = s*y_{t-1} + (1-s)*x_t, s = exp(-dt/max(tau,eps))
// x: (B,T,U) fp32. Memory-bound: 2 x 268MB => ~23us floor at 23.3 TB/s.
// Strategy: async global->LDS triple-buffered streaming (ASYNCcnt-tracked),
// per-thread recurrence state in a register, NT cache policy on the stream,
// coalesced nontemporal stores. 256 blocks x 2 tiles in flight ~= 16MB,
// matching the HBM bandwidth-delay product.

#define LP_DT 0.001f
constexpr int kB  = 32;
constexpr int kT  = 2048;
constexpr int kU  = 1024;
constexpr int US  = 128;              // units per block (1 unit per thread)
constexpr int TT  = 64;               // timesteps per LDS tile
constexpr int NT  = kT / TT;          // 32 tiles
constexpr int NBUF = 3;               // triple buffer
constexpr int TILE_FLOATS = TT * US;  // 8192 floats = 32KB per buffer
constexpr int NTHREADS = 128;         // 4 waves
constexpr int CHUNKS = TILE_FLOATS / 4;              // 2048 16B chunks per tile
constexpr int CHUNKS_PER_THREAD = CHUNKS / NTHREADS; // 16 async issues/wave/tile
constexpr int WAIT_N = 2 * CHUNKS_PER_THREAD;        // allow 2 newer tiles in flight

typedef int v4i __attribute__((vector_size(16)));

#if __has_builtin(__builtin_amdgcn_global_load_async_to_lds_b128)
#define HAVE_ASYNC_LDS 1
#else
#define HAVE_ASYNC_LDS 0
#endif

__device__ __forceinline__ void async_copy16(const float* gsrc, float* ldst) {
#if HAVE_ASYNC_LDS
  // global_load_async_to_lds_b128: DMA 16B/lane global -> LDS, ASYNCcnt-tracked.
  // cpol=1 -> TH_NT: single-use stream, don't pollute L2 (192MB < 268MB stream).
  __builtin_amdgcn_global_load_async_to_lds_b128(
      (v4i*)gsrc, (v4i*)ldst, /*offset=*/0, /*cpol=*/1);
#else
  // Fallback: synchronous 128-bit copy through VGPRs (compiler-managed waits).
  *reinterpret_cast<float4*>(ldst) = *reinterpret_cast<const float4*>(gsrc);
#endif
}

template <int N>
__device__ __forceinline__ void wait_asynccnt_le() {
#if HAVE_ASYNC_LDS
#if __has_builtin(__builtin_amdgcn_s_wait_asynccnt)
  __builtin_amdgcn_s_wait_asynccnt(N);
#else
  asm volatile("s_wait_asynccnt %0" ::"n"(N) : "memory");
#endif
#endif
}

__global__ __launch_bounds__(NTHREADS) void lowpass_scan_kernel(
    const float* __restrict__ x, const float* __restrict__ tau,
    const float* __restrict__ init, float* __restrict__ out) {
  __shared__ float lbuf[NBUF * TILE_FLOATS];  // 96KB of 320KB/WGP

  const int tid = threadIdx.x;
  const int u0  = blockIdx.x * US;
  const int b   = blockIdx.y;
  const int u   = u0 + tid;

  // Per-unit smoothing (one-time cost).
  const float tv  = tau[u];
  const float s   = expf(-LP_DT / fmaxf(tv, FLT_EPSILON));
  const float oms = 1.0f - s;
  float y = init[u];

  const float* xb = x + ((size_t)b * kT) * kU + u0;  // + t*kU + col
  float* ob = out + ((size_t)b * kT) * kU + u;       // + t*kU

  // Issue the async loads for one 64-row tile into LDS buffer `buf`.
  auto load_tile = [&](int tile, int buf) {
    const float* gt = xb + (size_t)tile * TT * kU;
    float* lb = lbuf + buf * TILE_FLOATS;
#pragma unroll
    for (int k = 0; k < CHUNKS_PER_THREAD; ++k) {
      const int c   = tid + k * NTHREADS;  // 16B-chunk index within the tile
      const int row = c >> 5;              // 32 chunks (512B) per row
      const int col = c & 31;
      async_copy16(gt + (size_t)row * kU + (size_t)col * 4, lb + (size_t)c * 4);
    }
  };

  // Prime the pipeline: three tiles in flight (48 outstanding/wave < 63 cap).
  load_tile(0, 0);
  load_tile(1, 1);
  load_tile(2, 2);

  for (int tile = 0; tile < NT; ++tile) {
    const int buf = tile % NBUF;
    // Allow the two newer tiles' loads to stay outstanding; wait for this tile's.
    wait_asynccnt_le<WAIT_N>();
    __syncthreads();  // all waves' portions of this tile are resident

    const float* lb = lbuf + buf * TILE_FLOATS + tid;
    float* orow = ob + (size_t)tile * TT * kU;
#pragma unroll 8
    for (int r = 0; r < TT; ++r) {
      const float xv = lb[(size_t)r * US];          // paired ds loads, conflict-free
      y = fmaf(s, y, oms * xv);                     // y = s*y + (1-s)*x
      __builtin_nontemporal_store(y, &orow[(size_t)r * kU]);  // coalesced NT store
    }

    __syncthreads();  // everyone done reading `buf` before refill
    if (tile + NBUF < NT) load_tile(tile + NBUF, buf);
  }
}

extern "C" void kernel_launch(void* const* d_in, const int* in_sizes, int n_in,
                              void* d_out, int out_size, void* d_ws, size_t ws_size,
                              hipStream_t stream) {
  const float* x    = (const float*)d_in[0];  // (B,T,U)
  const float* tau  = (const float*)d_in[1];  // (1,U)
  const float* init = (const float*)d_in[2];  // (1,U)
  float* out = (float*)d_out;                 // (B,T,U)

  dim3 grid(kU / US, kB);  // (8, 32) = 256 blocks
  lowpass_scan_kernel<<<grid, NTHREADS, 0, stream>>>(x, tau, init, out);
}